// TTTLinear_78091095376125
// MI455X (gfx1250) — compile-verified
//
#include <hip/hip_runtime.h>

typedef __attribute__((ext_vector_type(2))) float v2f;
typedef __attribute__((ext_vector_type(4))) float v4f;
typedef __attribute__((ext_vector_type(8))) float v8f;

#define B_TOK 8192
#define D_DIM 2048
#define H_DIM 64

// ---------------------------------------------------------------------------
// CDNA5 async global->LDS copy (GLOBAL_LOAD_ASYNC_TO_LDS_B128, ASYNCcnt).
// Builtin signature (from hipcc diagnostic): arg0 = v4i in AS1 (global),
// arg1 = v4i in AS3 (LDS), then imm offset, imm cpol.
// ---------------------------------------------------------------------------
#if __has_builtin(__builtin_amdgcn_global_load_async_to_lds_b128)
#define TTT_ASYNC 1
typedef int v4i_g __attribute__((vector_size(16)));
typedef __attribute__((address_space(1))) v4i_g as1_v4i;
typedef __attribute__((address_space(3))) v4i_g as3_v4i;
__device__ __forceinline__ void async_cp16(const float* g, float* l) {
  __builtin_amdgcn_global_load_async_to_lds_b128((as1_v4i*)g, (as3_v4i*)l, 0,
                                                 0);
}
#if __has_builtin(__builtin_amdgcn_s_wait_asynccnt)
#define ASYNC_WAIT(n) __builtin_amdgcn_s_wait_asynccnt(n)
#else
#define ASYNC_WAIT(n) asm volatile("s_wait_asynccnt %0" ::"n"(n))
#endif
#else
#define TTT_ASYNC 0
__device__ __forceinline__ void async_cp16(const float* g, float* l) {
  *(v4f*)l = *(const v4f*)g;
}
#define ASYNC_WAIT(n) ((void)0)
#endif

__device__ __forceinline__ v8f wmma4(v2f a, v2f b, v8f c) {
  // V_WMMA_F32_16X16X4_F32 : D(16x16 f32) = A(16x4 f32) * B(4x16 f32) + C
  return __builtin_amdgcn_wmma_f32_16x16x4_f32(false, a, false, b, (short)0, c,
                                               false, false);
}

__device__ __forceinline__ float wred32(float x) {
#pragma unroll
  for (int o = 16; o > 0; o >>= 1) x += __shfl_xor(x, o, 32);
  return x;
}

// ---------------------------------------------------------------------------
// Phase A: K = X@Wk+bk, V = X@Wv+bv, Q = X@Wq+bq, eta = sigmoid(X@w_lr+b_lr)
// GEMM M=8192, N=192, K=2048. Block = 4 waves, 64-row M tile.
// Wave owns all 64 rows x 48 cols: per k-step 4 A-frags + 3 B-frags, 12 WMMA.
// Kc=16 chunks, double-buffered async global->LDS staging.
// ---------------------------------------------------------------------------
#define KC 16
#define NCHUNK (D_DIM / KC)

__global__ __launch_bounds__(128) void ttt_proj_kernel(
    const float* __restrict__ x, const float* __restrict__ Wk,
    const float* __restrict__ bk, const float* __restrict__ Wv,
    const float* __restrict__ bv, const float* __restrict__ Wq,
    const float* __restrict__ bq, const float* __restrict__ w_lr,
    const float* __restrict__ b_lr, float* __restrict__ Ko,
    float* __restrict__ Vo, float* __restrict__ Qo, float* __restrict__ etao) {
  __shared__ float xs[2][64 * 20];   // [row][k], stride 20 (conflict-free)
  __shared__ float wt[2][KC * 192];  // [k][col] (Wk|Wv|Wq), stride 192

  const int t = threadIdx.x;
  const int lane = t & 31;
  const int wave = t >> 5;
  const int half = lane >> 4;  // 0: K=0,1  1: K=2,3
  const int l16 = lane & 15;
  const int m0 = blockIdx.x * 64;

  v8f acc[4][3];  // [m-group][n-tile]
  const v8f vz = {0.f, 0.f, 0.f, 0.f, 0.f, 0.f, 0.f, 0.f};
#pragma unroll
  for (int mg = 0; mg < 4; ++mg)
#pragma unroll
    for (int nt = 0; nt < 3; ++nt) acc[mg][nt] = vz;

  float spart = 0.0f;
  const int srow = t >> 1;
  const int scol0 = (t & 1) * 8;

  // ---- staging: 2 + 6 = 8 async b128 per thread per chunk ----
  auto stage = [&](int buf, int k0) {
#pragma unroll
    for (int i = 0; i < 2; ++i) {  // x tile: 64 rows x 16 k
      int lin = (i * 128 + t) * 4;
      int r = lin >> 4, c = lin & 15;
      async_cp16(&x[(m0 + r) * D_DIM + k0 + c], &xs[buf][r * 20 + c]);
    }
#pragma unroll
    for (int i = 0; i < 6; ++i) {  // weights: 16 k x 192 cols
      int lin = (i * 128 + t) * 4;
      int kc = lin / 192, col = lin % 192;
      const float* Wsrc = (col < 64) ? Wk : ((col < 128) ? Wv : Wq);
      async_cp16(&Wsrc[(k0 + kc) * H_DIM + (col & 63)],
                 &wt[buf][kc * 192 + col]);
    }
  };

  stage(0, 0);  // prologue

#pragma unroll 1
  for (int c = 0; c < NCHUNK; ++c) {
    const int buf = c & 1;
    if (c + 1 < NCHUNK) {
      stage(buf ^ 1, (c + 1) * KC);  // overlap next chunk's DMA with compute
      ASYNC_WAIT(8);                 // only the just-issued batch outstanding
    } else {
      ASYNC_WAIT(0);
    }
    __syncthreads();

    const float* xb = xs[buf];
    const float* wb = wt[buf];
#pragma unroll
    for (int k = 0; k < KC; k += 4) {
      v2f a[4];
#pragma unroll
      for (int mg = 0; mg < 4; ++mg)
        a[mg] = *(const v2f*)&xb[(mg * 16 + l16) * 20 + k + 2 * half];
#pragma unroll
      for (int nt = 0; nt < 3; ++nt) {
        int col = wave * 48 + nt * 16 + l16;
        v2f b;
        b.x = wb[(k + 2 * half) * 192 + col];
        b.y = wb[(k + 1 + 2 * half) * 192 + col];
#pragma unroll
        for (int mg = 0; mg < 4; ++mg)
          acc[mg][nt] = wmma4(a[mg], b, acc[mg][nt]);
      }
    }
    // learning-rate dot partial (thread pair per row)
    const int k0 = c * KC;
#pragma unroll
    for (int cc = 0; cc < 8; ++cc)
      spart += xb[srow * 20 + scol0 + cc] * w_lr[k0 + scol0 + cc];
    __syncthreads();  // protect buf before it is re-staged
  }

  float stot = spart + __shfl_xor(spart, 1, 32);
  if ((t & 1) == 0) {
    float s = stot + b_lr[0];
    etao[m0 + srow] = 1.0f / (1.0f + __expf(-s));
  }

#pragma unroll
  for (int nt = 0; nt < 3; ++nt) {
    const int cglob = wave * 48 + nt * 16;  // 0..176, multiple of 16
    const int mat = cglob >> 6;
    const int colm = (cglob & 63) + l16;
    const float* bias = (mat == 0) ? bk : ((mat == 1) ? bv : bq);
    float* dst = (mat == 0) ? Ko : ((mat == 1) ? Vo : Qo);
    float bb = bias[colm];
#pragma unroll
    for (int mg = 0; mg < 4; ++mg) {
#pragma unroll
      for (int g = 0; g < 8; ++g) {
        int row = m0 + mg * 16 + g + 8 * half;
        dst[row * H_DIM + colm] = acc[mg][nt][g] + bb;
      }
    }
  }
}

// ---------------------------------------------------------------------------
// Phase B: Hk = K @ W_init^T, Hq = Q @ W_init^T   (M=8192, N=64, K=64)
// ---------------------------------------------------------------------------
__global__ __launch_bounds__(128) void ttt_hidden_kernel(
    const float* __restrict__ Kw, const float* __restrict__ Qw,
    const float* __restrict__ Winit, float* __restrict__ Hk,
    float* __restrict__ Hq) {
  __shared__ float kt[64 * 68];
  __shared__ float qt[64 * 68];
  __shared__ float wl[64 * 68];  // W_init rows padded: wl[n*68+k]
  const int t = threadIdx.x;
  const int lane = t & 31;
  const int wave = t >> 5;
  const int half = lane >> 4;
  const int l16 = lane & 15;
  const int m0 = blockIdx.x * 64;

#pragma unroll
  for (int i = 0; i < 8; ++i) {
    int lin = (i * 128 + t) * 4;
    int r = lin >> 6, c = lin & 63;
    async_cp16(&Kw[(m0 + r) * H_DIM + c], &kt[r * 68 + c]);
    async_cp16(&Qw[(m0 + r) * H_DIM + c], &qt[r * 68 + c]);
    async_cp16(&Winit[r * H_DIM + c], &wl[r * 68 + c]);
  }
  ASYNC_WAIT(0);
  __syncthreads();

  const v8f vz = {0.f, 0.f, 0.f, 0.f, 0.f, 0.f, 0.f, 0.f};
  v8f ck[4], cq[4];
#pragma unroll
  for (int i = 0; i < 4; ++i) {
    ck[i] = vz;
    cq[i] = vz;
  }
  const int arow = wave * 16 + l16;
#pragma unroll
  for (int k = 0; k < 64; k += 4) {
    v2f ak = *(const v2f*)&kt[arow * 68 + k + 2 * half];
    v2f aq = *(const v2f*)&qt[arow * 68 + k + 2 * half];
#pragma unroll
    for (int nt = 0; nt < 4; ++nt) {
      v2f b = *(const v2f*)&wl[(nt * 16 + l16) * 68 + k + 2 * half];
      ck[nt] = wmma4(ak, b, ck[nt]);
      cq[nt] = wmma4(aq, b, cq[nt]);
    }
  }
#pragma unroll
  for (int nt = 0; nt < 4; ++nt) {
    int col = nt * 16 + l16;
#pragma unroll
    for (int g = 0; g < 8; ++g) {
      int row = m0 + wave * 16 + g + 8 * half;
      Hk[row * H_DIM + col] = ck[nt][g];
      Hq[row * H_DIM + col] = cq[nt][g];
    }
  }
}

// ---------------------------------------------------------------------------
// Phase C: per-token LN fwd/bwd, rank-1 update, W_new, ssl, Zpre.
// One wave32 per token; lane owns H-elements {2l, 2l+1}.
// ---------------------------------------------------------------------------
__global__ __launch_bounds__(256) void ttt_update_kernel(
    const float* __restrict__ Kw, const float* __restrict__ Vw,
    const float* __restrict__ Qw, const float* __restrict__ Hk,
    const float* __restrict__ Hq, const float* __restrict__ eta_in,
    const float* __restrict__ ln_g, const float* __restrict__ ln_b,
    const float* __restrict__ Winit, float* __restrict__ Wnew,
    float* __restrict__ ssl, float* __restrict__ Zpre) {
  __shared__ float wl[64 * 64];
  const int t = threadIdx.x;
  const int lane = t & 31;
  const int wave = t >> 5;
#pragma unroll
  for (int i = 0; i < 4; ++i) {
    int lin = (i * 256 + t) * 4;
    async_cp16(&Winit[lin], &wl[lin]);
  }
  ASYNC_WAIT(0);
  __syncthreads();

  const int tok = blockIdx.x * 8 + wave;
  const int j = lane * 2;
  v2f k2 = *(const v2f*)&Kw[tok * H_DIM + j];
  v2f v2 = *(const v2f*)&Vw[tok * H_DIM + j];
  v2f q2 = *(const v2f*)&Qw[tok * H_DIM + j];
  v2f hk = *(const v2f*)&Hk[tok * H_DIM + j];
  v2f hq = *(const v2f*)&Hq[tok * H_DIM + j];
  v2f g2 = *(const v2f*)&ln_g[j];
  v2f bb = *(const v2f*)&ln_b[j];
  float eta = eta_in[tok];

  const float invH = 1.0f / 64.0f;
  // LN forward on hk
  float m1 = wred32(hk.x + hk.y) * invH;
  v2f d1 = hk - m1;
  float var1 = wred32(d1.x * d1.x + d1.y * d1.y) * invH;
  float rs1 = rsqrtf(var1 + 1e-6f);
  v2f hat1 = d1 * rs1;
  v2f r = k2 + hat1 * g2 + bb - v2;  // pred - v
  float sl = wred32(r.x * r.x + r.y * r.y) * invH;
  // LN backward -> gh = dL/dh
  v2f dy = r * (2.0f * invH);
  v2f dh = dy * g2;
  float a1 = wred32(dh.x + dh.y) * invH;
  float a2 = wred32(dh.x * hat1.x + dh.y * hat1.y) * invH;
  v2f gh = (dh - a1 - hat1 * a2) * rs1;

  // h2 = W_new q = Hq - eta*(k.q)*gh, then LN -> Zpre
  float kq = wred32(k2.x * q2.x + k2.y * q2.y);
  v2f h2 = hq - gh * (eta * kq);
  float m2 = wred32(h2.x + h2.y) * invH;
  v2f d2 = h2 - m2;
  float var2 = wred32(d2.x * d2.x + d2.y * d2.y) * invH;
  float rs2 = rsqrtf(var2 + 1e-6f);
  v2f zp = q2 + (d2 * rs2) * g2 + bb;
  *(v2f*)&Zpre[tok * H_DIM + j] = zp;

  if (lane == 0) ssl[tok] = sl;

  // W_new = W_init - eta * gh k^T  (rank-1), coalesced float2 rows
  float* wout = &Wnew[(size_t)tok * (H_DIM * H_DIM)];
#pragma unroll
  for (int i = 0; i < 64; ++i) {
    float ghi = __shfl((i & 1) ? gh.y : gh.x, i >> 1, 32);
    v2f w2 = *(const v2f*)&wl[i * 64 + j];
    v2f o = w2 - k2 * (eta * ghi);
    *(v2f*)&wout[i * 64 + j] = o;
  }
}

// ---------------------------------------------------------------------------
// Phase D: z = Zpre(8192x64) @ Wo(64x2048) + bo
// Block tile 64x128, 4 waves. Wave owns all 64 rows x 32 cols:
// per k-step 4 A-frags + 2 B-frags, 8 WMMA.
// ---------------------------------------------------------------------------
__global__ __launch_bounds__(128) void ttt_out_kernel(
    const float* __restrict__ Zpre, const float* __restrict__ Wo,
    const float* __restrict__ bo, float* __restrict__ z) {
  __shared__ float zt[64 * 68];
  __shared__ float wo_s[64 * 128];  // [k][nl], stride 128
  const int t = threadIdx.x;
  const int lane = t & 31;
  const int wave = t >> 5;
  const int half = lane >> 4;
  const int l16 = lane & 15;
  const int m0 = (int)(blockIdx.x >> 4) * 64;
  const int n0 = (int)(blockIdx.x & 15) * 128;

#pragma unroll
  for (int i = 0; i < 8; ++i) {
    int lin = (i * 128 + t) * 4;
    int r = lin >> 6, c = lin & 63;
    async_cp16(&Zpre[(m0 + r) * H_DIM + c], &zt[r * 68 + c]);
  }
#pragma unroll
  for (int i = 0; i < 16; ++i) {
    int lin = (i * 128 + t) * 4;
    int k = lin >> 7, nl = lin & 127;
    async_cp16(&Wo[k * D_DIM + n0 + nl], &wo_s[k * 128 + nl]);
  }
  ASYNC_WAIT(0);
  __syncthreads();

  const v8f vz = {0.f, 0.f, 0.f, 0.f, 0.f, 0.f, 0.f, 0.f};
  v8f acc[4][2];
#pragma unroll
  for (int mg = 0; mg < 4; ++mg)
#pragma unroll
    for (int nt = 0; nt < 2; ++nt) acc[mg][nt] = vz;

#pragma unroll
  for (int k = 0; k < 64; k += 4) {
    v2f a[4];
#pragma unroll
    for (int mg = 0; mg < 4; ++mg)
      a[mg] = *(const v2f*)&zt[(mg * 16 + l16) * 68 + k + 2 * half];
#pragma unroll
    for (int nt = 0; nt < 2; ++nt) {
      int nl = wave * 32 + nt * 16 + l16;
      v2f b;
      b.x = wo_s[(k + 2 * half) * 128 + nl];
      b.y = wo_s[(k + 1 + 2 * half) * 128 + nl];
#pragma unroll
      for (int mg = 0; mg < 4; ++mg)
        acc[mg][nt] = wmma4(a[mg], b, acc[mg][nt]);
    }
  }
#pragma unroll
  for (int nt = 0; nt < 2; ++nt) {
    int col = n0 + wave * 32 + nt * 16 + l16;
    float bb = bo[col];
#pragma unroll
    for (int mg = 0; mg < 4; ++mg) {
#pragma unroll
      for (int g = 0; g < 8; ++g) {
        int row = m0 + mg * 16 + g + 8 * half;
        z[(size_t)row * D_DIM + col] = acc[mg][nt][g] + bb;
      }
    }
  }
}

// ---------------------------------------------------------------------------
extern "C" void kernel_launch(void* const* d_in, const int* in_sizes, int n_in,
                              void* d_out, int out_size, void* d_ws,
                              size_t ws_size, hipStream_t stream) {
  const float* x = (const float*)d_in[0];
  const float* Wk = (const float*)d_in[1];
  const float* bk = (const float*)d_in[2];
  const float* Wv = (const float*)d_in[3];
  const float* bv = (const float*)d_in[4];
  const float* Wq = (const float*)d_in[5];
  const float* bq = (const float*)d_in[6];
  const float* Wo = (const float*)d_in[7];
  const float* bo = (const float*)d_in[8];
  const float* lng = (const float*)d_in[9];
  const float* lnb = (const float*)d_in[10];
  const float* wlr = (const float*)d_in[11];
  const float* blr = (const float*)d_in[12];
  const float* Wini = (const float*)d_in[13];

  float* out = (float*)d_out;
  float* z = out;                                     // [B, D]
  float* Wnew = out + (size_t)B_TOK * D_DIM;          // [B, H, H]
  float* ssl = Wnew + (size_t)B_TOK * H_DIM * H_DIM;  // [B]

  float* ws = (float*)d_ws;
  const size_t BH = (size_t)B_TOK * H_DIM;
  float* Kw = ws;
  float* Vw = ws + BH;
  float* Qw = ws + 2 * BH;
  float* Hk = ws + 3 * BH;
  float* Hq = ws + 4 * BH;
  float* Zp = ws + 5 * BH;
  float* eta = ws + 6 * BH;

  hipLaunchKernelGGL(ttt_proj_kernel, dim3(B_TOK / 64), dim3(128), 0, stream,
                     x, Wk, bk, Wv, bv, Wq, bq, wlr, blr, Kw, Vw, Qw, eta);
  hipLaunchKernelGGL(ttt_hidden_kernel, dim3(B_TOK / 64), dim3(128), 0, stream,
                     Kw, Qw, Wini, Hk, Hq);
  hipLaunchKernelGGL(ttt_update_kernel, dim3(B_TOK / 8), dim3(256), 0, stream,
                     Kw, Vw, Qw, Hk, Hq, eta, lng, lnb, Wini, Wnew, ssl, Zp);
  hipLaunchKernelGGL(ttt_out_kernel, dim3((B_TOK / 64) * (D_DIM / 128)),
                     dim3(128), 0, stream, Zp, Wo, bo, z);
}